// SecureOptimizedBlockReLU_85890755985457
// MI455X (gfx1250) — compile-verified
//
#include <hip/hip_runtime.h>

typedef float v4f __attribute__((ext_vector_type(4)));

// Tensor: (N=16, C=64, H=256, W=256) fp32, NCHW contiguous.
// Groups: ch 0-15  -> 1x1 blocks (== ReLU)
//         ch 16-39 -> 2x2 blocks
//         ch 40-55 -> 3x3 blocks (H,W padded 256->258 with zeros)
//         ch 56-63 -> passthrough
//
// Memory-bound: 256 MiB in + 256 MiB out, single touch. Design targets the
// 23.3 TB/s HBM roofline: b128 coalesced traffic, NT cache hints (512 MB
// stream > 192 MB L2), async global->LDS DMA for the cross-row 3x3 group.

// ---------------------------------------------------------------------------
// Kernel A: channels 0-15 (ReLU) + 56-63 (copy).
// grid = (64, 24, 16): x = 256-float4 strip, y = channel idx, z = batch.
// ---------------------------------------------------------------------------
__global__ __launch_bounds__(256) void blk1_copy_kernel(const float* __restrict__ in,
                                                        float* __restrict__ out) {
    unsigned q  = blockIdx.x * 256u + threadIdx.x;     // float4 index in plane
    unsigned ci = blockIdx.y;                          // 0..23
    unsigned n  = blockIdx.z;
    unsigned ch = (ci < 16u) ? ci : (ci + 40u);        // 0..15 or 56..63
    size_t base = ((size_t)(n * 64u + ch) << 16) + ((size_t)q << 2);

    v4f v = __builtin_nontemporal_load((const v4f*)(in + base));
    if (ci < 16u) {  // scalar (block-uniform) branch
        v.x = (v.x >= 0.f) ? v.x : 0.f;
        v.y = (v.y >= 0.f) ? v.y : 0.f;
        v.z = (v.z >= 0.f) ? v.z : 0.f;
        v.w = (v.w >= 0.f) ? v.w : 0.f;
    }
    __builtin_nontemporal_store(v, (v4f*)(out + base));
}

// ---------------------------------------------------------------------------
// Kernel B: channels 16-39, 2x2 blocks. Each lane owns 2 rows x 4 cols
// (two 2x2 blocks): two b128 loads + two b128 stores, coalesced along W.
// grid = (32, 24, 16): x = (row-pair, col-quad) strip, y = channel, z = batch.
// ---------------------------------------------------------------------------
__global__ __launch_bounds__(256) void blk2_kernel(const float* __restrict__ in,
                                                   float* __restrict__ out) {
    unsigned t  = blockIdx.x * 256u + threadIdx.x;     // 0..8191
    unsigned cq = t & 63u;                             // column quad
    unsigned br = t >> 6;                              // row pair 0..127
    unsigned c  = blockIdx.y + 16u;                    // 16..39
    unsigned n  = blockIdx.z;
    size_t base = ((size_t)(n * 64u + c) << 16)
                + (size_t)(2u * br) * 256u + (size_t)cq * 4u;

    v4f a = __builtin_nontemporal_load((const v4f*)(in + base));          // row 2br
    v4f b = __builtin_nontemporal_load((const v4f*)(in + base + 256u));   // row 2br+1

    float s0 = (a.x + a.y) + (b.x + b.y);
    float s1 = (a.z + a.w) + (b.z + b.w);
    if (s0 < 0.f) { a.x = 0.f; a.y = 0.f; b.x = 0.f; b.y = 0.f; }
    if (s1 < 0.f) { a.z = 0.f; a.w = 0.f; b.z = 0.f; b.w = 0.f; }

    __builtin_nontemporal_store(a, (v4f*)(out + base));
    __builtin_nontemporal_store(b, (v4f*)(out + base + 256u));
}

// ---------------------------------------------------------------------------
// Kernel C: channels 40-55, 3x3 blocks (86 block-rows per plane, last one
// clipped at row 255 == zero padding). One workgroup per (plane, block-row).
// 3x256 row band is DMA'd to LDS via CDNA5 async b128 loads (ASYNCcnt),
// 86 lanes form the block sums, 192 lanes apply masks and store b128 NT.
// grid = (86, 16, 16): x = block-row, y = channel-40, z = batch.
// ---------------------------------------------------------------------------
__global__ __launch_bounds__(256) void blk3_kernel(const float* __restrict__ in,
                                                   float* __restrict__ out) {
    __shared__ __align__(16) float smem[768 + 96];   // 3x256 band + 86 masks

    unsigned t  = threadIdx.x;
    unsigned br = blockIdx.x;           // block-row 0..85
    unsigned c  = blockIdx.y + 40u;     // 40..55
    unsigned n  = blockIdx.z;
    unsigned r0 = 3u * br;              // first row of band (<= 255)

    size_t band_base = ((size_t)(n * 64u + c) << 16) + (size_t)r0 * 256u;
    const float* src = in  + band_base;
    float*       dst = out + band_base;

    unsigned lds0 = (unsigned)(size_t)(&smem[0]);    // LDS byte address of smem

    // Async DMA: 192 lanes each bring one float4 of the 3x256 band into LDS.
    if (t < 192u) {
        unsigned i    = t >> 6;          // row within band, 0..2
        unsigned col4 = t & 63u;         // float4 column
        if (r0 + i < 256u) {
            unsigned goff = i * 1024u + col4 * 16u;   // byte offset from src
            unsigned loff = lds0 + t * 16u;           // LDS byte address
            asm volatile("global_load_async_to_lds_b128 %0, %1, %2"
                         :
                         : "v"(loff), "v"(goff),
                           "s"((unsigned long long)(size_t)src)
                         : "memory");
        }
    }
    asm volatile("s_wait_asynccnt 0" ::: "memory");
    __syncthreads();

    unsigned rmax = (r0 + 3u <= 256u) ? 3u : (256u - r0);

    // 86 lanes compute the 3x3 (edge-clipped) block sums -> sign masks.
    if (t < 86u) {
        unsigned c0   = 3u * t;
        unsigned cmax = (c0 + 3u <= 256u) ? 3u : (256u - c0);
        float s = 0.f;
        for (unsigned i = 0; i < rmax; ++i)
            for (unsigned k = 0; k < cmax; ++k)
                s += smem[i * 256u + c0 + k];
        smem[768u + t] = (s >= 0.f) ? 1.f : 0.f;
    }
    __syncthreads();

    // Apply mask and stream the band out: one b128 NT store per lane.
    if (t < 192u) {
        unsigned i    = t >> 6;
        unsigned col4 = t & 63u;
        if (r0 + i < 256u) {
            unsigned cb = col4 * 4u;
            v4f v = *(const v4f*)&smem[i * 256u + cb];   // ds_load_b128
            v.x *= smem[768u + (cb + 0u) / 3u];
            v.y *= smem[768u + (cb + 1u) / 3u];
            v.z *= smem[768u + (cb + 2u) / 3u];
            v.w *= smem[768u + (cb + 3u) / 3u];
            __builtin_nontemporal_store(v, (v4f*)(dst + (size_t)i * 256u + cb));
        }
    }
}

// ---------------------------------------------------------------------------
extern "C" void kernel_launch(void* const* d_in, const int* in_sizes, int n_in,
                              void* d_out, int out_size, void* d_ws, size_t ws_size,
                              hipStream_t stream) {
    const float* in = (const float*)d_in[0];
    float* out      = (float*)d_out;
    (void)in_sizes; (void)n_in; (void)out_size; (void)d_ws; (void)ws_size;

    blk1_copy_kernel<<<dim3(64, 24, 16), 256, 0, stream>>>(in, out);
    blk2_kernel    <<<dim3(32, 24, 16), 256, 0, stream>>>(in, out);
    blk3_kernel    <<<dim3(86, 16, 16), 256, 0, stream>>>(in, out);
}